// SO3ConvolutionInvariants_20650202759258
// MI455X (gfx1250) — compile-verified
//
#include <hip/hip_runtime.h>
#include <stdint.h>

// SO3 invariants: out[e,l] = (1/sqrt(2l+1)) * dot(x[e, seg_l], y[e, seg_l])
// FEAT=24 (segments 3,5,7,9), N=2M edges. Pure streaming problem:
// 416 MB traffic -> ~18us floor at 23.3 TB/s. Strategy: async-stage tiles to
// LDS with contiguous b128 bursts (ASYNCcnt path), elementwise product,
// contract 24->4 via chained v_wmma_f32_16x16x4_f32 against a constant
// block-diagonal scale matrix, then emit one coalesced b128 store per lane.

#define FEAT   24
#define STRIDE 28            // padded LDS row stride (floats): 112B, 16B-aligned
#define ROWS   32            // edges per wave tile
#define WAVES  4             // waves per block
#define BLOCK  (WAVES * 32)  // 128 threads, 128 edges per block

typedef float f4  __attribute__((ext_vector_type(4)));
typedef float v2f __attribute__((ext_vector_type(2)));
typedef float v8f __attribute__((ext_vector_type(8)));
typedef int   i4  __attribute__((ext_vector_type(4)));

// ---------- CDNA5 async global->LDS path (ASYNCcnt) ----------------------
__device__ __forceinline__ void async_load_b128(const float* g, float* l) {
#if defined(__gfx1250__) && __has_builtin(__builtin_amdgcn_global_load_async_to_lds_b128)
  // Signature (from clang diagnostic): (int4 AS1* src, int4 AS3* dst, imm, imm)
  __builtin_amdgcn_global_load_async_to_lds_b128(
      (__attribute__((address_space(1))) i4*)(g),
      (__attribute__((address_space(3))) i4*)(l), 0, 0);
#else
  asm volatile("global_load_async_to_lds_b128 %0, %1, off"
               :: "v"((uint32_t)(uintptr_t)l), "v"(g) : "memory");
#endif
}

__device__ __forceinline__ void wait_async0() {
#if defined(__gfx1250__) && __has_builtin(__builtin_amdgcn_s_wait_asynccnt)
  __builtin_amdgcn_s_wait_asynccnt(0);
#else
  asm volatile("s_wait_asynccnt 0" ::: "memory");
#endif
}

__global__ __launch_bounds__(BLOCK) void so3_inv_kernel(
    const float* __restrict__ X, const float* __restrict__ Y,
    float* __restrict__ O, int nEdges) {
  __shared__ float lX[WAVES][ROWS * STRIDE];
  __shared__ float lY[WAVES][ROWS * STRIDE];
  __shared__ float lZ[WAVES][ROWS * STRIDE];
  __shared__ float lO[WAVES][ROWS * 4];      // out tile: 32 rows x 4 invariants

  const int lane  = threadIdx.x & 31;
  const int w     = threadIdx.x >> 5;
  const int tile0 = (blockIdx.x * WAVES + w) * ROWS;
  if (tile0 >= nEdges) return;
  int valid = nEdges - tile0; if (valid > ROWS) valid = ROWS;
  const int maxChunk = valid * 6;  // 6 x 16B chunks per 24-float row

  const float* gx = X + (size_t)tile0 * FEAT;
  const float* gy = Y + (size_t)tile0 * FEAT;
  float* lx = lX[w]; float* ly = lY[w]; float* lz = lZ[w]; float* lo = lO[w];

  // Stage 32 rows x 96B of X and Y: 6 async b128 instructions per input,
  // each moving a fully contiguous 512B burst (f enumerates memory linearly).
#pragma unroll
  for (int i = 0; i < 6; ++i) {
    int f   = i * 32 + lane;
    int fc  = (f < maxChunk) ? f : 0;        // clamp partial tiles (benign dup)
    int row = fc / 6, c = fc - row * 6;
    int loff = row * STRIDE + c * 4;
    async_load_b128(gx + (size_t)fc * 4, &lx[loff]);
    async_load_b128(gy + (size_t)fc * 4, &ly[loff]);
  }
  wait_async0();
  __builtin_amdgcn_wave_barrier();

  // Elementwise product z = x*y for this lane's own row (2-way LDS conflicts max).
  f4 zv[6];
#pragma unroll
  for (int i = 0; i < 6; ++i) {
    f4 xv = *(const f4*)&lx[lane * STRIDE + i * 4];
    f4 yv = *(const f4*)&ly[lane * STRIDE + i * 4];
    zv[i] = xv * yv;
  }

#if defined(__gfx1250__) && __has_builtin(__builtin_amdgcn_wmma_f32_16x16x4_f32)
  // ---- WMMA contraction: D(16x16) = A(16x4 of z) x B(4x16 of W), 6 K-steps.
  // Write z to LDS so it can be re-read in the WMMA A striping
  // (lane half selects K-pair; same-wave DS ops are in-order, no barrier).
#pragma unroll
  for (int i = 0; i < 6; ++i)
    *(f4*)&lz[lane * STRIDE + i * 4] = zv[i];
  __builtin_amdgcn_wave_barrier();
  asm volatile("" ::: "memory");

  const int nCol  = lane & 15;         // B/C/D column owned by this lane
  const int khalf = (lane >> 4) << 1;  // lane half -> K offset {0,2}

  // Constant B: B[k][n] = 1/sqrt(size_n) iff k falls in segment n (n<4), else 0.
  v2f B[6];
#pragma unroll
  for (int s = 0; s < 6; ++s) {
#pragma unroll
    for (int j = 0; j < 2; ++j) {
      int k   = s * 4 + khalf + j;
      int seg = (k < 3) ? 0 : (k < 8) ? 1 : (k < 15) ? 2 : 3;
      float wv = (seg == 0) ? 0.57735026919f :
                 (seg == 1) ? 0.44721359550f :
                 (seg == 2) ? 0.37796447301f : 0.33333333333f;
      B[s][j] = (nCol == seg) ? wv : 0.0f;
    }
  }

#pragma unroll
  for (int t = 0; t < 2; ++t) {        // two 16-edge tiles per wave
    v8f acc = {};
    const int rbase = t * 16 + nCol;   // A row owned by this lane (M = lane%16)
#pragma unroll
    for (int s = 0; s < 6; ++s) {
      v2f a = *(const v2f*)&lz[rbase * STRIDE + s * 4 + khalf]; // conflict-free b64
      acc = __builtin_amdgcn_wmma_f32_16x16x4_f32(
          /*neg_a=*/false, a, /*neg_b=*/false, B[s],
          /*c_mod=*/(short)0, acc, /*reuse_a=*/false, /*reuse_b=*/false);
    }
    // D layout: VGPR v holds (M = v + 8*(lane>=16), N = lane%16). Only N<4 valid.
    // Scatter D into the LDS out-tile with constant offsets (pairs into
    // ds_store_2addr_b32), then store coalesced below.
    if (nCol < 4) {
      const int mb = t * 16 + ((lane >> 4) << 3);
      float* dst = &lo[mb * 4 + nCol];
#pragma unroll
      for (int v = 0; v < 8; ++v) dst[v * 4] = acc[v];
    }
  }
  __builtin_amdgcn_wave_barrier();
  asm volatile("" ::: "memory");

  // One contiguous 512B burst per wave: lane stores its own row's 4 invariants.
  if (tile0 + lane < nEdges) {
    f4 o = *(const f4*)&lo[lane * 4];
    *(f4*)&O[(size_t)(tile0 + lane) * 4] = o;
  }
#else
  // ---- VALU fallback: segmented dot from registers, coalesced b128 store.
  float d0 = zv[0][0] + zv[0][1] + zv[0][2];
  float d1 = zv[0][3] + zv[1][0] + zv[1][1] + zv[1][2] + zv[1][3];
  float d2 = zv[2][0] + zv[2][1] + zv[2][2] + zv[2][3] +
             zv[3][0] + zv[3][1] + zv[3][2];
  float d3 = zv[3][3] + zv[4][0] + zv[4][1] + zv[4][2] + zv[4][3] +
             zv[5][0] + zv[5][1] + zv[5][2] + zv[5][3];
  int e = tile0 + lane;
  if (e < nEdges) {
    f4 o = {d0 * 0.57735026919f, d1 * 0.44721359550f,
            d2 * 0.37796447301f, d3 * 0.33333333333f};
    *(f4*)&O[(size_t)e * 4] = o;
  }
#endif
}

extern "C" void kernel_launch(void* const* d_in, const int* in_sizes, int n_in,
                              void* d_out, int out_size, void* d_ws, size_t ws_size,
                              hipStream_t stream) {
  const float* X = (const float*)d_in[0];
  const float* Y = (const float*)d_in[1];
  float* O = (float*)d_out;
  const int nEdges = in_sizes[0] / FEAT;           // 2,000,000
  const int blocks = (nEdges + BLOCK - 1) / BLOCK; // 15,625 for 2M
  so3_inv_kernel<<<blocks, BLOCK, 0, stream>>>(X, Y, O, nEdges);
}